// TransducerLoss_78984448573900
// MI455X (gfx1250) — compile-verified
//
#include <hip/hip_runtime.h>
#include <hip/hip_bf16.h>
#include <stdint.h>

#define NEG_INF (-1e30f)

typedef unsigned int v4u __attribute__((ext_vector_type(4)));
typedef int          v8i __attribute__((ext_vector_type(8)));
typedef int          v4i __attribute__((ext_vector_type(4)));

// Problem dims are fixed by the reference: B=4, T=256, U1=101, V=1024.
constexpr int TT         = 256;
constexpr int U1         = 101;
constexpr int UMAX       = 100;
constexpr int VV         = 1024;
constexpr int CHUNKS     = 8;
constexpr int CHUNK_T    = TT / CHUNKS;     // 32 t-rows per TDM chunk
constexpr int CHUNK_ROWS = CHUNK_T * U1;    // 3232 flattened (t,u) rows per chunk
constexpr int PLANE      = TT * U1;         // 25856 blank values per batch

// log(exp(a)+exp(b)); cheap form: exp arg <= 0 so 1+e is in [1,2] where
// plain v_log is plenty accurate. Safe for the -1e30 log-zero sentinel.
__device__ __forceinline__ float lae(float a, float b) {
    float m = fmaxf(a, b);
    float d = fabsf(a - b);
    return m + __logf(1.0f + __expf(-d));
}

// Issue one TDM descriptor: DMA chunk k of the blank plane (stride-V column
// of logits) into LDS. D# bitfields per CDNA5 ISA sect. 8.3-8.5:
//   2D tensor, data_size=4B, tensor_dim0=1, tile_dim0=1,
//   tensor_dim1=tile_dim1=CHUNK_ROWS, tensor_dim0_stride=V.
__device__ __forceinline__ void tdm_load_blank_chunk(const float* gbase, uint32_t lds_byte, int k)
{
    unsigned long long ga = (unsigned long long)(uintptr_t)gbase
                          + (unsigned long long)k * (unsigned long long)CHUNK_ROWS * VV * 4ull;
    uint32_t la = lds_byte + (uint32_t)k * (uint32_t)CHUNK_ROWS * 4u;

    v4u g0;
    g0[0] = 1u;                                                   // count=1 (valid), user mode
    g0[1] = la;                                                   // lds_addr (bytes)
    g0[2] = (uint32_t)(ga & 0xFFFFFFFFull);                       // global_addr[31:0]
    g0[3] = (uint32_t)((ga >> 32) & 0x01FFFFFFull) | (2u << 30);  // global_addr[56:32] | type=2

    v8i g1;
    g1[0] = (2 << 16);                                 // data_size=4B; no multicast/pad/iterate
    g1[1] = (1 << 16);                                 // tensor_dim0 = 1
    g1[2] = (CHUNK_ROWS & 0xFFFF) << 16;               // tensor_dim1[15:0]
    g1[3] = ((CHUNK_ROWS >> 16) & 0xFFFF) | (1 << 16); // tensor_dim1[31:16] | tile_dim0=1
    g1[4] = CHUNK_ROWS;                                // tile_dim1 (tile_dim2=0)
    g1[5] = VV;                                        // tensor_dim0_stride[31:0] (elements)
    g1[6] = 0;
    g1[7] = 0;

    v4i g2 = {0, 0, 0, 0};
    v4i g3 = {0, 0, 0, 0};

#if __has_include(<hip/amd_detail/amd_gfx1250_TDM.h>)
    v8i g4 = {0, 0, 0, 0, 0, 0, 0, 0};
    __builtin_amdgcn_tensor_load_to_lds(g0, g1, g2, g3, g4, 0);   // clang-23 6-arg form
#else
    __builtin_amdgcn_tensor_load_to_lds(g0, g1, g2, g3, 0);       // ROCm 7.2 5-arg form
#endif
}

// Chunks complete in issue order; chunk j done <=> TENSORcnt <= CHUNKS-1-j.
__device__ __forceinline__ void wait_chunk_done(int j) {
    switch (j) {
        case 0:  __builtin_amdgcn_s_wait_tensorcnt((short)7); break;
        case 1:  __builtin_amdgcn_s_wait_tensorcnt((short)6); break;
        case 2:  __builtin_amdgcn_s_wait_tensorcnt((short)5); break;
        case 3:  __builtin_amdgcn_s_wait_tensorcnt((short)4); break;
        case 4:  __builtin_amdgcn_s_wait_tensorcnt((short)3); break;
        case 5:  __builtin_amdgcn_s_wait_tensorcnt((short)2); break;
        case 6:  __builtin_amdgcn_s_wait_tensorcnt((short)1); break;
        default: __builtin_amdgcn_s_wait_tensorcnt((short)0); break;
    }
}

// One wave (32 lanes) per batch. Each lane owns u = 4*lane .. 4*lane+3.
__global__ __launch_bounds__(32)
void rnnt_alpha_scan_tdm_kernel(const float* __restrict__ logits,
                                const int*   __restrict__ labels,
                                const int*   __restrict__ tlen,
                                const int*   __restrict__ ulen,
                                float*       __restrict__ loss)   // ws: one float per batch
{
    __shared__ float sBlank[PLANE];            // blank[t][u] staged by the TDM

    const int b    = blockIdx.x;
    const int lane = threadIdx.x;
    const int u0   = lane * 4;
    const float* __restrict__ base = logits + (size_t)b * (size_t)PLANE * (size_t)VV;

    // Kick off the full blank-plane DMA (8 in-flight descriptors).
    const uint32_t lds0 = (uint32_t)(uintptr_t)&sBlank[0];
#pragma unroll
    for (int k = 0; k < CHUNKS; ++k) tdm_load_blank_chunk(base, lds0, k);

    // Per-lane gather offsets for label logits c[t][u] = logits[t,u-1,labels[u-1]].
    bool   bvalid[4], cvalid[4];
    size_t coff[4];
#pragma unroll
    for (int i = 0; i < 4; ++i) {
        int u = u0 + i;
        bvalid[i] = (u < U1);
        cvalid[i] = (u >= 1 && u < U1);
        int li    = cvalid[i] ? labels[(size_t)b * UMAX + (u - 1)] : 0;
        coff[i]   = (size_t)(cvalid[i] ? (u - 1) : 0) * (size_t)VV + (size_t)li;
    }

    const int tlast = tlen[b] - 1;
    const int ufin  = ulen[b];

    // alpha0 = [0, -inf, ...]
    float alpha[4];
#pragma unroll
    for (int i = 0; i < 4; ++i) alpha[i] = (u0 + i == 0) ? 0.0f : NEG_INF;

    // Software-pipelined label gathers.
    float ccur[4], cnxt[4];
    {
        const float* rp = base;                // t = 0
#pragma unroll
        for (int i = 0; i < 4; ++i) ccur[i] = cvalid[i] ? rp[coff[i]] : 0.0f;
    }

    for (int j = 0; j < CHUNKS; ++j) {
        wait_chunk_done(j);                    // blank rows [32j, 32j+32) now in LDS
        for (int tt = 0; tt < CHUNK_T; ++tt) {
            const int t = j * CHUNK_T + tt;

            // Warm L2 for label gathers ~8 rows ahead (global_prefetch_b8).
            if (t + 8 < TT) {
                const float* pp = base + (size_t)(t + 8) * U1 * VV;
#pragma unroll
                for (int i = 0; i < 4; ++i)
                    if (cvalid[i]) __builtin_prefetch(pp + coff[i], 0, 0);
            }
            // Issue next-row label gathers; consumed after this step's compute.
            {
                const int tn = (t + 1 < TT) ? (t + 1) : t;
                const float* rp = base + (size_t)tn * U1 * VV;
#pragma unroll
                for (int i = 0; i < 4; ++i) cnxt[i] = cvalid[i] ? rp[coff[i]] : 0.0f;
            }

            // blank[t-1,u] from LDS (zero row at t==0).
            float bp[4];
            if (t == 0) {
                bp[0] = bp[1] = bp[2] = bp[3] = 0.0f;
            } else {
                const float* sp = &sBlank[(size_t)(t - 1) * U1];
#pragma unroll
                for (int i = 0; i < 4; ++i) bp[i] = bvalid[i] ? sp[u0 + i] : 0.0f;
            }

            // ---- C = inclusive cumsum_u(c); padded lanes contribute 0 ----
            float s0 = ccur[0];
            float s1 = s0 + ccur[1];
            float s2 = s1 + ccur[2];
            float s3 = s2 + ccur[3];
            float run = s3;
#pragma unroll
            for (int d = 1; d < 32; d <<= 1) {
                float o = __shfl_up(run, d, 32);
                if (lane >= d) run += o;
            }
            float eadd = __shfl_up(run, 1, 32);
            if (lane == 0) eadd = 0.0f;
            float C[4] = {eadd + s0, eadd + s1, eadd + s2, eadd + s3};

            // ---- x = alpha_prev + blank_prev - C; inclusive prefix-logaddexp ----
            float x[4];
#pragma unroll
            for (int i = 0; i < 4; ++i)
                x[i] = bvalid[i] ? (alpha[i] + bp[i] - C[i]) : NEG_INF;
            float y0 = x[0];
            float y1 = lae(y0, x[1]);
            float y2 = lae(y1, x[2]);
            float y3 = lae(y2, x[3]);
            float r = y3;
#pragma unroll
            for (int d = 1; d < 32; d <<= 1) {
                float o = __shfl_up(r, d, 32);
                if (lane >= d) r = lae(r, o);
            }
            float elog = __shfl_up(r, 1, 32);
            if (lane == 0) elog = NEG_INF;
            float yv[4] = {y0, y1, y2, y3};
#pragma unroll
            for (int i = 0; i < 4; ++i) alpha[i] = C[i] + lae(elog, yv[i]);

            // ---- record loss for this batch ----
            if (t == tlast) {
#pragma unroll
                for (int i = 0; i < 4; ++i)
                    if (u0 + i == ufin) loss[b] = -alpha[i];
            }

            // rotate label pipeline
#pragma unroll
            for (int i = 0; i < 4; ++i) ccur[i] = cnxt[i];
        }
    }
}

__global__ void rnnt_finalize_kernel(const float* __restrict__ loss,
                                     float* __restrict__ out, int B)
{
    if (blockIdx.x == 0 && threadIdx.x == 0) {
        float s = 0.0f;
        for (int b = 0; b < B; ++b) s += loss[b];
        out[0] = s / (float)B;
    }
}

extern "C" void kernel_launch(void* const* d_in, const int* in_sizes, int n_in,
                              void* d_out, int out_size, void* d_ws, size_t ws_size,
                              hipStream_t stream) {
    const float* logits = (const float*)d_in[0];
    const int*   labels = (const int*)d_in[1];
    const int*   tlen   = (const int*)d_in[2];
    const int*   ulen   = (const int*)d_in[3];

    const int B = in_sizes[2];                 // logit_lengths has B entries

    float* ws  = (float*)d_ws;                 // B per-batch losses
    float* out = (float*)d_out;

    rnnt_alpha_scan_tdm_kernel<<<B, 32, 0, stream>>>(logits, labels, tlen, ulen, ws);
    rnnt_finalize_kernel<<<1, 1, 0, stream>>>(ws, out, B);
}